// RHOEncoder_49469433316012
// MI455X (gfx1250) — compile-verified
//
#include <hip/hip_runtime.h>
#include <hip/hip_bf16.h>

#define N_NODES 100000
#define D       128
#define N_EDGES 1600000

typedef __attribute__((ext_vector_type(2))) float v2f;
typedef __attribute__((ext_vector_type(8))) float v8f;

// ---------------- degree / dinv ----------------
__global__ void k_deg_init(int* deg) {
    int i = blockIdx.x * blockDim.x + threadIdx.x;
    if (i < N_NODES) deg[i] = 1;   // self-loop contributes 1
}

__global__ void k_deg_count(const int* __restrict__ dst, int* deg) {
    int e = blockIdx.x * blockDim.x + threadIdx.x;
    if (e < N_EDGES) atomicAdd(&deg[dst[e]], 1);
}

__global__ void k_dinv(const int* __restrict__ deg, float* __restrict__ dinv) {
    int i = blockIdx.x * blockDim.x + threadIdx.x;
    if (i < N_NODES) dinv[i] = rsqrtf((float)deg[i]);   // deg >= 1 always
}

// ---------------- dense GEMM: C = A(100000x128) @ W(128x128), fp32 WMMA ----------------
__global__ void k_gemm_wmma(const float* __restrict__ A, const float* __restrict__ W,
                            float* __restrict__ C) {
    const int tile_m = blockIdx.x;          // 0..6249  (16 rows each)
    const int wave   = threadIdx.x >> 5;    // 0..7 -> column tile (16 cols each)
    const int lane   = threadIdx.x & 31;
    const int mlane  = lane & 15;
    const int khalf  = (lane >> 4) << 1;    // 0 for lanes 0-15, 2 for lanes 16-31

    const int row     = tile_m * 16 + mlane;
    const int colbase = wave * 16;
    const int col     = colbase + mlane;

    v8f acc = {};
    const float* arow = A + (size_t)row * D;
#pragma unroll
    for (int k = 0; k < D; k += 4) {
        // A 16x4 frag: VGPR0 = A[row][k+khalf], VGPR1 = A[row][k+khalf+1]
        v2f a = *(const v2f*)(arow + k + khalf);
        // B 4x16 frag: VGPR0 = W[k+khalf][col], VGPR1 = W[k+khalf+1][col]
        v2f b;
        b.x = W[(k + khalf) * D + col];
        b.y = W[(k + khalf + 1) * D + col];
        acc = __builtin_amdgcn_wmma_f32_16x16x4_f32(false, a, false, b,
                                                    (short)0, acc, false, false);
    }
    // C layout: VGPR j -> (M = j + 8*(lane>=16), N = lane&15)
    const int mhi = (lane >> 4) << 3;
#pragma unroll
    for (int j = 0; j < 8; ++j) {
        int m = tile_m * 16 + j + mhi;
        C[(size_t)m * D + colbase + mlane] = acc[j];
    }
}

// ---------------- spmm1: out1 = dinv2*XW (self-loop) then edge atomics ----------------
__global__ void k_spmm1_init(const float* __restrict__ XW, const float* __restrict__ dinv,
                             float* __restrict__ out1) {
    int i4 = blockIdx.x * blockDim.x + threadIdx.x;   // float4 index
    if (i4 >= N_NODES * (D / 4)) return;
    int node = i4 >> 5;                               // D/4 == 32 float4 per row
    float dv = dinv[node];
    float s  = dv * dv;
    float4 v = ((const float4*)XW)[i4];
    float4 r = make_float4(s * v.x, s * v.y, s * v.z, s * v.w);
    ((float4*)out1)[i4] = r;
}

__global__ void k_spmm1_edges(const int* __restrict__ src, const int* __restrict__ dst,
                              const float* __restrict__ dinv, const float* __restrict__ X,
                              float* out) {
    int gid  = blockIdx.x * blockDim.x + threadIdx.x;
    int e    = gid >> 5;
    int lane = gid & 31;
    if (e >= N_EDGES) return;
    int s = src[e], d = dst[e];
    float w = dinv[s] * dinv[d];
    float4 v = ((const float4*)(X + (size_t)d * D))[lane];
    float* orow = out + (size_t)s * D + lane * 4;
    atomicAdd(orow + 0, w * v.x);
    atomicAdd(orow + 1, w * v.y);
    atomicAdd(orow + 2, w * v.z);
    atomicAdd(orow + 3, w * v.w);
}

// ---------------- base_h = relu(out1 + b)  (in place);  Hs = base_h * K_cw ----------------
__global__ void k_relu_hs(float* __restrict__ bh_inout, const float* __restrict__ b,
                          const float* __restrict__ Kcw, float* __restrict__ Hs) {
    int i4 = blockIdx.x * blockDim.x + threadIdx.x;
    if (i4 >= N_NODES * (D / 4)) return;
    int c4 = i4 & 31;
    float4 v  = ((const float4*)bh_inout)[i4];
    float4 bb = ((const float4*)b)[c4];
    float4 kk = ((const float4*)Kcw)[c4];
    float4 r;
    r.x = fmaxf(v.x + bb.x, 0.0f);
    r.y = fmaxf(v.y + bb.y, 0.0f);
    r.z = fmaxf(v.z + bb.z, 0.0f);
    r.w = fmaxf(v.w + bb.w, 0.0f);
    ((float4*)bh_inout)[i4] = r;
    float4 hs = make_float4(r.x * kk.x, r.y * kk.y, r.z * kk.z, r.w * kk.w);
    ((float4*)Hs)[i4] = hs;
}

// final = 0.5*((2-k)*bh - Hs + k*S2 + S3), S2=spmm(bh), S3=spmm(Hs)
// init:  d_out = 0.5*((2-k+k*dinv2)*bh + (dinv2-1)*Hs)
__global__ void k_final_init(const float* __restrict__ bh, const float* __restrict__ Hs,
                             const float* __restrict__ dinv, const float* __restrict__ kcc,
                             float* __restrict__ out) {
    int i4 = blockIdx.x * blockDim.x + threadIdx.x;
    if (i4 >= N_NODES * (D / 4)) return;
    int node = i4 >> 5;
    float dv = dinv[node];
    float d2 = dv * dv;
    float k  = kcc[0];
    float ca = 0.5f * (2.0f - k + k * d2);
    float cb = 0.5f * (d2 - 1.0f);
    float4 vb = ((const float4*)bh)[i4];
    float4 vh = ((const float4*)Hs)[i4];
    float4 r;
    r.x = ca * vb.x + cb * vh.x;
    r.y = ca * vb.y + cb * vh.y;
    r.z = ca * vb.z + cb * vh.z;
    r.w = ca * vb.w + cb * vh.w;
    ((float4*)out)[i4] = r;
}

// edges: d_out[src] += 0.5*w*(k*bh[dst] + Hs[dst])
__global__ void k_final_edges(const int* __restrict__ src, const int* __restrict__ dst,
                              const float* __restrict__ dinv, const float* __restrict__ bh,
                              const float* __restrict__ Hs, const float* __restrict__ kcc,
                              float* out) {
    int gid  = blockIdx.x * blockDim.x + threadIdx.x;
    int e    = gid >> 5;
    int lane = gid & 31;
    if (e >= N_EDGES) return;
    int s = src[e], d = dst[e];
    float k = kcc[0];
    float w = 0.5f * dinv[s] * dinv[d];
    float4 vb = ((const float4*)(bh + (size_t)d * D))[lane];
    float4 vh = ((const float4*)(Hs + (size_t)d * D))[lane];
    float* orow = out + (size_t)s * D + lane * 4;
    atomicAdd(orow + 0, w * (k * vb.x + vh.x));
    atomicAdd(orow + 1, w * (k * vb.y + vh.y));
    atomicAdd(orow + 2, w * (k * vb.z + vh.z));
    atomicAdd(orow + 3, w * (k * vb.w + vh.w));
}

extern "C" void kernel_launch(void* const* d_in, const int* in_sizes, int n_in,
                              void* d_out, int out_size, void* d_ws, size_t ws_size,
                              hipStream_t stream) {
    const float* h     = (const float*)d_in[0];
    const int*   ei    = (const int*)d_in[1];
    const float* W     = (const float*)d_in[2];
    const float* bg    = (const float*)d_in[3];
    const float* kcc   = (const float*)d_in[4];
    const float* Kcw   = (const float*)d_in[5];
    float* out = (float*)d_out;

    const int* src = ei;
    const int* dst = ei + N_EDGES;

    // workspace layout
    size_t nd = (size_t)N_NODES * D;                 // 12.8M floats
    char* ws  = (char*)d_ws;
    float* bufA = (float*)ws;                        // XW, later reused as Hs
    float* bufB = (float*)(ws + nd * sizeof(float)); // out1 -> base_h (in place)
    int*   deg  = (int*)  (ws + 2 * nd * sizeof(float));
    float* dinv = (float*)(ws + 2 * nd * sizeof(float) + N_NODES * sizeof(int));

    const int T = 256;
    const int gN    = (N_NODES + T - 1) / T;
    const int gE    = (N_EDGES + T - 1) / T;
    const int gND4  = (int)((nd / 4 + T - 1) / T);     // 12500
    const int gEdgeWave = (int)(((size_t)N_EDGES * 32 + T - 1) / T); // 200000
    const int gGemm = N_NODES / 16;                    // 6250

    // 1) degrees & dinv
    k_deg_init <<<gN, T, 0, stream>>>(deg);
    k_deg_count<<<gE, T, 0, stream>>>(dst, deg);
    k_dinv     <<<gN, T, 0, stream>>>(deg, dinv);

    // 2) XW = h @ W   (WMMA f32)
    k_gemm_wmma<<<gGemm, T, 0, stream>>>(h, W, bufA);

    // 3) spmm(XW) -> bufB
    k_spmm1_init <<<gND4, T, 0, stream>>>(bufA, dinv, bufB);
    k_spmm1_edges<<<gEdgeWave, T, 0, stream>>>(src, dst, dinv, bufA, bufB);

    // 4) base_h = relu(bufB + b) in place; Hs = base_h * K_cw -> bufA
    k_relu_hs<<<gND4, T, 0, stream>>>(bufB, bg, Kcw, bufA);

    // 5) fused final: init + single edge pass accumulating both spmms into d_out
    k_final_init <<<gND4, T, 0, stream>>>(bufB, bufA, dinv, kcc, out);
    k_final_edges<<<gEdgeWave, T, 0, stream>>>(src, dst, dinv, bufB, bufA, kcc, out);
}